// CrossRegionAttention_31301721653561
// MI455X (gfx1250) — compile-verified
//
#include <hip/hip_runtime.h>
#include <math.h>

// ---------------------------------------------------------------------------
// CrossRegionAttention on MI455X (gfx1250), fp32 end-to-end.
// Memory-bound op (~630MB @ 23.3TB/s => ~27us floor; fp32 WMMA compute ~2us),
// so we keep exact fp32 numerics and use V_WMMA_F32_16X16X4_F32 for all GEMMs.
// ---------------------------------------------------------------------------

typedef float v2f __attribute__((ext_vector_type(2)));
typedef float v8f __attribute__((ext_vector_type(8)));

#define DIM    2048
#define HEADS  16
#define HD     128
#define NB     128      // batch
#define NR     49       // image regions
#define NS     100      // text tokens
#define TI     (NB*NR)  // 6272  (multiple of 128)
#define TT     (NB*NS)  // 12800 (multiple of 128)

// ----------------------------- reductions ----------------------------------

__device__ __forceinline__ float block_sum(float v, float* sh) {
  #pragma unroll
  for (int off = 16; off > 0; off >>= 1) v += __shfl_xor(v, off, 32);
  const int wid = threadIdx.x >> 5, lane = threadIdx.x & 31;
  if (lane == 0) sh[wid] = v;
  __syncthreads();
  float t = (lane < 8) ? sh[lane] : 0.f;
  #pragma unroll
  for (int off = 16; off > 0; off >>= 1) t += __shfl_xor(t, off, 32);
  __syncthreads();              // allow sh reuse by a following call
  return t;
}

// ------------------------------- LayerNorm ---------------------------------

__global__ __launch_bounds__(256)
void ln_kernel(const float* __restrict__ x, const float* __restrict__ g,
               const float* __restrict__ b, float* __restrict__ y) {
  __shared__ float sh[8];
  const size_t t = blockIdx.x;
  const float* row = x + t * DIM;
  float lv[8];
  float s = 0.f;
  #pragma unroll
  for (int i = 0; i < 8; ++i) { lv[i] = row[threadIdx.x + i * 256]; s += lv[i]; }
  s = block_sum(s, sh);
  const float mean = s * (1.f / DIM);
  float vs = 0.f;
  #pragma unroll
  for (int i = 0; i < 8; ++i) { float d = lv[i] - mean; vs += d * d; }
  vs = block_sum(vs, sh);
  const float rstd = rsqrtf(vs * (1.f / DIM) + 1e-5f);
  #pragma unroll
  for (int i = 0; i < 8; ++i) {
    const int c = threadIdx.x + i * 256;
    y[t * DIM + c] = (lv[i] - mean) * rstd * g[c] + b[c];
  }
}

// ------------------------- fp32 WMMA tiled GEMM ----------------------------
// C[t,n] = act( sum_k A(t,k) * W[k,n] + bias[n] ),  A(t,k) = k<2048 ? A0 : A1
// Tile: BM=128, BN=64, BK=16. 256 threads = 8 waves; wave = 2x2 subtiles of
// 16x16, K stepped 4 at a time through V_WMMA_F32_16X16X4_F32.

template <int ACT>
__global__ __launch_bounds__(256)
void gemm_wmma_kernel(const float* __restrict__ A0, const float* __restrict__ A1,
                      const float* __restrict__ W,  const float* __restrict__ bias,
                      float* __restrict__ C, int K) {
  constexpr int AS_LD = 132;   // 128 + pad
  constexpr int BS_LD = 68;    // 64 + pad
  __shared__ float As[16 * AS_LD];
  __shared__ float Bs[16 * BS_LD];

  const int tid  = threadIdx.x;
  const int t0   = blockIdx.y * 128;
  const int n0   = blockIdx.x * 64;
  const int wv   = tid >> 5;
  const int lane = tid & 31;
  const int half = lane >> 4;    // 0: lanes 0-15, 1: lanes 16-31
  const int l16  = lane & 15;
  const int waveM = (wv & 3) * 32;
  const int waveN = (wv >> 2) * 32;

  v8f acc[2][2];
  #pragma unroll
  for (int mi = 0; mi < 2; ++mi)
    #pragma unroll
    for (int ni = 0; ni < 2; ++ni) acc[mi][ni] = (v8f){};

  for (int kc = 0; kc < K; kc += 16) {
    const float* Asrc = (kc < DIM) ? A0 : A1;
    const int    koff = (kc < DIM) ? kc : kc - DIM;

    // A tile: 128 rows x 16 k, float4 loads (2 per thread), store k-major.
    #pragma unroll
    for (int i = 0; i < 2; ++i) {
      const int idx = tid + i * 256;
      const int row = idx >> 2;
      const int kq  = (idx & 3) << 2;
      const float4 v = *(const float4*)(&Asrc[(size_t)(t0 + row) * DIM + koff + kq]);
      As[(kq + 0) * AS_LD + row] = v.x;
      As[(kq + 1) * AS_LD + row] = v.y;
      As[(kq + 2) * AS_LD + row] = v.z;
      As[(kq + 3) * AS_LD + row] = v.w;
    }
    // B tile: 16 k x 64 n, one float4 per thread.
    {
      const int row = tid >> 4;
      const int cq  = (tid & 15) << 2;
      const float4 v = *(const float4*)(&W[(size_t)(kc + row) * DIM + n0 + cq]);
      Bs[row * BS_LD + cq + 0] = v.x;
      Bs[row * BS_LD + cq + 1] = v.y;
      Bs[row * BS_LD + cq + 2] = v.z;
      Bs[row * BS_LD + cq + 3] = v.w;
    }
    __syncthreads();

    #pragma unroll
    for (int ks = 0; ks < 4; ++ks) {
      // A 16x4 layout: lanes 0-15 -> K=k0,k0+1 ; lanes 16-31 -> K=k0+2,k0+3
      const int kb = ks * 4 + half * 2;
      v2f a[2], b[2];
      #pragma unroll
      for (int mi = 0; mi < 2; ++mi) {
        const int m = waveM + mi * 16 + l16;
        a[mi].x = As[(kb + 0) * AS_LD + m];
        a[mi].y = As[(kb + 1) * AS_LD + m];
      }
      #pragma unroll
      for (int ni = 0; ni < 2; ++ni) {
        const int n = waveN + ni * 16 + l16;
        b[ni].x = Bs[(kb + 0) * BS_LD + n];
        b[ni].y = Bs[(kb + 1) * BS_LD + n];
      }
      #pragma unroll
      for (int mi = 0; mi < 2; ++mi)
        #pragma unroll
        for (int ni = 0; ni < 2; ++ni)
          acc[mi][ni] = __builtin_amdgcn_wmma_f32_16x16x4_f32(
              false, a[mi], false, b[ni], (short)0, acc[mi][ni], false, false);
    }
    __syncthreads();
  }

  // C/D layout: VGPR j -> M=j (lanes 0-15), M=8+j (lanes 16-31); N = lane&15.
  #pragma unroll
  for (int mi = 0; mi < 2; ++mi) {
    #pragma unroll
    for (int ni = 0; ni < 2; ++ni) {
      const int gcol = n0 + waveN + ni * 16 + l16;
      const float bval = bias[gcol];
      #pragma unroll
      for (int j = 0; j < 8; ++j) {
        const int grow = t0 + waveM + mi * 16 + half * 8 + j;
        float v = acc[mi][ni][j] + bval;
        if (ACT == 1) v = fmaxf(v, 0.f);
        C[(size_t)grow * DIM + gcol] = v;
      }
    }
  }
}

// ------------------------------ attention ----------------------------------
// One block per (b,h). K tile (then V tile) staged in LDS (<=100*128 f32).
// probs written to d_out (they ARE outputs), softmaxed in place, then P*V.

__global__ __launch_bounds__(256)
void attn_kernel(const float* __restrict__ Qb, const float* __restrict__ Kb,
                 const float* __restrict__ Vb, const float* __restrict__ bias,
                 const float* __restrict__ temp, float* __restrict__ probs,
                 float* __restrict__ attout, int nq, int nk) {
  __shared__ float kv[NS * HD];            // 51.2 KB, max(nk)=100
  const int bh = blockIdx.x;
  const int b = bh / HEADS, h = bh % HEADS;
  const int tid = threadIdx.x;
  const float inv = 1.f / fmaxf(temp[0], 0.01f);
  float* prow = probs + (size_t)bh * nq * nk;

  for (int i = tid; i < nk * HD; i += 256) {
    const int k = i >> 7, e = i & 127;
    kv[i] = Kb[((size_t)(b * nk + k)) * DIM + h * HD + e];
  }
  __syncthreads();

  // scores: it = QK^T/scale + bias[key]
  for (int i = tid; i < nq * nk; i += 256) {
    const int q = i / nk, k = i - q * nk;
    const float4* qp = (const float4*)(Qb + ((size_t)(b * nq + q)) * DIM + h * HD);
    const float4* kp = (const float4*)(&kv[k * HD]);
    float s = 0.f;
    #pragma unroll 8
    for (int e = 0; e < HD / 4; ++e) {
      const float4 a = qp[e], c = kp[e];
      s += a.x * c.x + a.y * c.y + a.z * c.z + a.w * c.w;
    }
    prow[i] = s * inv + bias[k];
  }
  __threadfence_block();
  __syncthreads();

  // softmax: one wave per row
  const int wv = tid >> 5, lane = tid & 31;
  for (int q = wv; q < nq; q += 8) {
    float* r = prow + (size_t)q * nk;
    float v[4];
    float m = -3.4e38f;
    int cnt = 0;
    for (int k = lane; k < nk; k += 32) { v[cnt] = r[k]; m = fmaxf(m, v[cnt]); ++cnt; }
    #pragma unroll
    for (int off = 16; off > 0; off >>= 1) m = fmaxf(m, __shfl_xor(m, off, 32));
    float sum = 0.f;
    for (int c2 = 0; c2 < cnt; ++c2) { v[c2] = __expf(v[c2] - m); sum += v[c2]; }
    #pragma unroll
    for (int off = 16; off > 0; off >>= 1) sum += __shfl_xor(sum, off, 32);
    const float isum = 1.f / sum;
    cnt = 0;
    for (int k = lane; k < nk; k += 32) { r[k] = v[cnt] * isum; ++cnt; }
  }
  __threadfence_block();
  __syncthreads();

  // V into LDS (overwrite K tile)
  for (int i = tid; i < nk * HD; i += 256) {
    const int k = i >> 7, e = i & 127;
    kv[i] = Vb[((size_t)(b * nk + k)) * DIM + h * HD + e];
  }
  __syncthreads();

  // att = P @ V  -> [b, q, h, hd] flat == rows of [T, DIM]
  for (int i = tid; i < nq * HD; i += 256) {
    const int q = i >> 7, e = i & 127;
    const float* pr = prow + (size_t)q * nk;
    float s = 0.f;
    for (int k = 0; k < nk; ++k) s += pr[k] * kv[k * HD + e];
    attout[((size_t)(b * nq + q)) * DIM + h * HD + e] = s;
  }
}

// ---------------------- gate LN + sigmoid + combine ------------------------
// enh[t,:] += sigmoid(LN(gate_pre[t,:])) * outp[t,:]   (enh holds relu residual)

__global__ __launch_bounds__(256)
void gate_combine_kernel(const float* __restrict__ gp, const float* __restrict__ lng,
                         const float* __restrict__ lnb, const float* __restrict__ outp,
                         float* __restrict__ enh) {
  __shared__ float sh[8];
  const size_t t = blockIdx.x;
  const float* row = gp + t * DIM;
  float lv[8];
  float s = 0.f;
  #pragma unroll
  for (int i = 0; i < 8; ++i) { lv[i] = row[threadIdx.x + i * 256]; s += lv[i]; }
  s = block_sum(s, sh);
  const float mean = s * (1.f / DIM);
  float vs = 0.f;
  #pragma unroll
  for (int i = 0; i < 8; ++i) { float d = lv[i] - mean; vs += d * d; }
  vs = block_sum(vs, sh);
  const float rstd = rsqrtf(vs * (1.f / DIM) + 1e-5f);
  #pragma unroll
  for (int i = 0; i < 8; ++i) {
    const int c = threadIdx.x + i * 256;
    const float ln = (lv[i] - mean) * rstd * lng[c] + lnb[c];
    const float g = 1.f / (1.f + __expf(-ln));
    enh[t * DIM + c] += g * outp[t * DIM + c];
  }
}

// ------------------------------- launcher ----------------------------------

extern "C" void kernel_launch(void* const* d_in, const int* in_sizes, int n_in,
                              void* d_out, int out_size, void* d_ws, size_t ws_size,
                              hipStream_t stream) {
  (void)in_sizes; (void)n_in; (void)out_size; (void)ws_size;
  const float* img   = (const float*)d_in[0];
  const float* txt   = (const float*)d_in[1];
  const float* img_q_w = (const float*)d_in[2];   const float* img_q_b = (const float*)d_in[3];
  const float* img_k_w = (const float*)d_in[4];   const float* img_k_b = (const float*)d_in[5];
  const float* img_v_w = (const float*)d_in[6];   const float* img_v_b = (const float*)d_in[7];
  const float* txt_q_w = (const float*)d_in[8];   const float* txt_q_b = (const float*)d_in[9];
  const float* txt_k_w = (const float*)d_in[10];  const float* txt_k_b = (const float*)d_in[11];
  const float* txt_v_w = (const float*)d_in[12];  const float* txt_v_b = (const float*)d_in[13];
  const float* img_o_w = (const float*)d_in[14];  const float* img_o_b = (const float*)d_in[15];
  const float* txt_o_w = (const float*)d_in[16];  const float* txt_o_b = (const float*)d_in[17];
  const float* img_r_w = (const float*)d_in[18];  const float* img_r_b = (const float*)d_in[19];
  const float* txt_r_w = (const float*)d_in[20];  const float* txt_r_b = (const float*)d_in[21];
  const float* img_g_w = (const float*)d_in[22];  const float* img_g_b = (const float*)d_in[23];
  const float* img_g_lg = (const float*)d_in[24]; const float* img_g_lb = (const float*)d_in[25];
  const float* txt_g_w = (const float*)d_in[26];  const float* txt_g_b = (const float*)d_in[27];
  const float* txt_g_lg = (const float*)d_in[28]; const float* txt_g_lb = (const float*)d_in[29];
  const float* img_n_g = (const float*)d_in[30];  const float* img_n_b = (const float*)d_in[31];
  const float* txt_n_g = (const float*)d_in[32];  const float* txt_n_b = (const float*)d_in[33];
  const float* temperature = (const float*)d_in[34];
  const float* region_pos  = (const float*)d_in[35];  // [R]
  const float* token_pos   = (const float*)d_in[36];  // [S]

  const size_t IMG = (size_t)TI * DIM;   // 12,845,056
  const size_t TXT = (size_t)TT * DIM;   // 26,214,400

  float* ws = (float*)d_ws;
  float* img_n   = ws;
  float* txt_n   = img_n + IMG;
  float* iq      = txt_n + TXT;
  float* ik      = iq + IMG;
  float* iv      = ik + IMG;
  float* tq      = iv + IMG;
  float* tk      = tq + TXT;
  float* tv      = tk + TXT;
  float* img_att = tv + TXT;
  float* txt_att = img_att + IMG;
  // reuse after attention / projections:
  float* img_outp = iq;
  float* txt_outp = tq;
  float* img_gate = img_n;
  float* txt_gate = txt_n;

  float* out    = (float*)d_out;
  float* enh_im = out;
  float* enh_tx = out + IMG;
  float* it_p   = enh_tx + TXT;
  float* ti_p   = it_p + (size_t)NB * HEADS * NR * NS;

  const dim3 blk(256);
  const dim3 gI(DIM / 64, TI / 128);   // 32 x 49
  const dim3 gT(DIM / 64, TT / 128);   // 32 x 100

  // 1) LayerNorms
  ln_kernel<<<TI, blk, 0, stream>>>(img, img_n_g, img_n_b, img_n);
  ln_kernel<<<TT, blk, 0, stream>>>(txt, txt_n_g, txt_n_b, txt_n);

  // 2) q/k/v projections (K=2048)
  gemm_wmma_kernel<0><<<gI, blk, 0, stream>>>(img_n, img_n, img_q_w, img_q_b, iq, DIM);
  gemm_wmma_kernel<0><<<gI, blk, 0, stream>>>(img_n, img_n, img_k_w, img_k_b, ik, DIM);
  gemm_wmma_kernel<0><<<gI, blk, 0, stream>>>(img_n, img_n, img_v_w, img_v_b, iv, DIM);
  gemm_wmma_kernel<0><<<gT, blk, 0, stream>>>(txt_n, txt_n, txt_q_w, txt_q_b, tq, DIM);
  gemm_wmma_kernel<0><<<gT, blk, 0, stream>>>(txt_n, txt_n, txt_k_w, txt_k_b, tk, DIM);
  gemm_wmma_kernel<0><<<gT, blk, 0, stream>>>(txt_n, txt_n, txt_v_w, txt_v_b, tv, DIM);

  // 3) cross attention (img->text and text->img)
  attn_kernel<<<NB * HEADS, blk, 0, stream>>>(iq, tk, tv, token_pos, temperature,
                                              it_p, img_att, NR, NS);
  attn_kernel<<<NB * HEADS, blk, 0, stream>>>(tq, ik, iv, region_pos, temperature,
                                              ti_p, txt_att, NS, NR);

  // 4) output projections (att @ out_w + b) -> reuse q buffers
  gemm_wmma_kernel<0><<<gI, blk, 0, stream>>>(img_att, img_att, img_o_w, img_o_b, img_outp, DIM);
  gemm_wmma_kernel<0><<<gT, blk, 0, stream>>>(txt_att, txt_att, txt_o_w, txt_o_b, txt_outp, DIM);

  // 5) gate pre-activations: concat(x, att) @ gate_w + b  (K=4096) -> reuse norm buffers
  gemm_wmma_kernel<0><<<gI, blk, 0, stream>>>(img, img_att, img_g_w, img_g_b, img_gate, 2 * DIM);
  gemm_wmma_kernel<0><<<gT, blk, 0, stream>>>(txt, txt_att, txt_g_w, txt_g_b, txt_gate, 2 * DIM);

  // 6) residual path: relu(x @ Wr + b) straight into d_out
  gemm_wmma_kernel<1><<<gI, blk, 0, stream>>>(img, img, img_r_w, img_r_b, enh_im, DIM);
  gemm_wmma_kernel<1><<<gT, blk, 0, stream>>>(txt, txt, txt_r_w, txt_r_b, enh_tx, DIM);

  // 7) enh += sigmoid(LN(gate)) * outp
  gate_combine_kernel<<<TI, blk, 0, stream>>>(img_gate, img_g_lg, img_g_lb, img_outp, enh_im);
  gate_combine_kernel<<<TT, blk, 0, stream>>>(txt_gate, txt_g_lg, txt_g_lb, txt_outp, enh_tx);
}